// SeerAttnQwen2DecoderLayer_29661044146336
// MI455X (gfx1250) — compile-verified
//
#include <hip/hip_runtime.h>
#include <hip/hip_bf16.h>
#include <math.h>

#define S_LEN 2048
#define HID   2048
#define HQ    16
#define HKV   4
#define HD    128
#define BS    64
#define QB    (S_LEN / BS)   // 32
#define GH    128
#define INTER 5632
#define GRP   (HQ / HKV)     // 4
#define THR_F 0.004f
#define EPS_F 1e-6f
#define NEG_F (-1e30f)

// GEMM macro-tile
#define BM 128
#define BN 64
#define BK 16

typedef __attribute__((ext_vector_type(2))) float v2f;
typedef __attribute__((ext_vector_type(8))) float v8f;

// ---------------------------------------------------------------------------
// CDNA5 fp32 WMMA: D = A(16x4) * B(4x16) + C(16x16), wave32.
// ---------------------------------------------------------------------------
__device__ __forceinline__ v8f wmma4_f32(v2f a, v2f b, v8f c) {
  return __builtin_amdgcn_wmma_f32_16x16x4_f32(
      /*neg_a=*/false, a, /*neg_b=*/false, b,
      /*c_mod=*/(short)0, c, /*reuse_a=*/false, /*reuse_b=*/false);
}

// ---------------------------------------------------------------------------
// CDNA5 async global->LDS copy (ASYNCcnt-tracked), 16 bytes per lane.
// LDS destination is the 32-bit LDS byte offset (low half of the generic
// address of a __shared__ object).
// ---------------------------------------------------------------------------
__device__ __forceinline__ unsigned lds_off(const void* p) {
  return (unsigned)(uintptr_t)p;
}
__device__ __forceinline__ void async_b128(unsigned lds_byte_off, const float* g) {
  asm volatile("global_load_async_to_lds_b128 %0, %1, off"
               :: "v"(lds_byte_off), "v"(g)
               : "memory");
}
__device__ __forceinline__ void wait_asynccnt_0() {
  asm volatile("s_wait_asynccnt 0x0" ::: "memory");
}
__device__ __forceinline__ void wait_asynccnt_3() {
  asm volatile("s_wait_asynccnt 0x3" ::: "memory");
}

// ---------------------------------------------------------------------------
// fp32 WMMA GEMM: C[M,N] = A[M,K] @ B[K,N] (+bias[N]) (+residual[M,N]).
// 256 threads = 8 waves; macro-tile BM=128 x BN=64, K in BK=16 slabs,
// double-buffered in LDS via async global->LDS loads. Wave w computes rows
// [BM*by + 16w, +16) x all 64 cols (4 accumulators sharing one A fragment).
// A fragment (ISA 7.12.2, 32-bit A 16x4): lane%16 = M, lane/16 -> K-pair.
// B mirrors with lane%16 = N. C: vgpr r -> M = r + 8*(lane/16).
// ---------------------------------------------------------------------------
__global__ __launch_bounds__(256) void gemm_wmma_f32_kernel(
    const float* __restrict__ A, const float* __restrict__ B,
    const float* __restrict__ bias, const float* __restrict__ residual,
    float* __restrict__ C, int M, int N, int K) {
  __shared__ float Atile[2][BM * BK];
  __shared__ float Btile[2][BK * BN];
  const int tid  = threadIdx.x;
  const int wave = tid >> 5;
  const int lane = tid & 31;
  const int half = lane >> 4;
  const int col  = lane & 15;
  const int mbase = blockIdx.y * BM;
  const int nbase = blockIdx.x * BN;

  const unsigned aoff[2] = {lds_off(&Atile[0][0]), lds_off(&Atile[1][0])};
  const unsigned boff[2] = {lds_off(&Btile[0][0]), lds_off(&Btile[1][0])};

  // Per-thread staging: A tile = 2048 floats (2 x float4/thread),
  // B tile = 1024 floats (1 x float4/thread). All 16B aligned.
  const int ae0 = tid * 4;
  const int ae1 = ae0 + 1024;
  const int be0 = tid * 4;

#define GEMM_STAGE(kb_, buf_)                                                 \
  do {                                                                        \
    const int k0_ = (kb_) * BK;                                               \
    async_b128(aoff[buf_] + ae0 * 4,                                          \
               A + (size_t)(mbase + (ae0 >> 4)) * K + k0_ + (ae0 & 15));      \
    async_b128(aoff[buf_] + ae1 * 4,                                          \
               A + (size_t)(mbase + (ae1 >> 4)) * K + k0_ + (ae1 & 15));      \
    async_b128(boff[buf_] + be0 * 4,                                          \
               B + (size_t)(k0_ + (be0 >> 6)) * N + nbase + (be0 & 63));      \
  } while (0)

  v8f zero = {};
  v8f acc[4] = {zero, zero, zero, zero};

  const int nkb = K / BK;
  GEMM_STAGE(0, 0);
  for (int kb = 0; kb < nkb; ++kb) {
    const int buf = kb & 1;
    if (kb + 1 < nkb) {
      GEMM_STAGE(kb + 1, buf ^ 1);  // next slab in flight
      wait_asynccnt_3();            // current slab landed
    } else {
      wait_asynccnt_0();
    }
    __syncthreads();
    const float* At = &Atile[buf][(wave * 16 + col) * BK];
    const float* Bt = &Btile[buf][0];
#pragma unroll
    for (int kk = 0; kk < BK; kk += 4) {
      v2f a;
      a.x = At[kk + 2 * half];
      a.y = At[kk + 2 * half + 1];
#pragma unroll
      for (int j = 0; j < 4; ++j) {
        v2f b;
        b.x = Bt[(kk + 2 * half) * BN + j * 16 + col];
        b.y = Bt[(kk + 2 * half + 1) * BN + j * 16 + col];
        acc[j] = wmma4_f32(a, b, acc[j]);
      }
    }
    __syncthreads();  // done reading buf before it is overwritten
  }
#undef GEMM_STAGE

  const int m0 = mbase + wave * 16;
#pragma unroll
  for (int j = 0; j < 4; ++j) {
    const int n = nbase + j * 16 + col;
    const float bval = bias ? bias[n] : 0.0f;
#pragma unroll
    for (int r = 0; r < 8; ++r) {
      const int m = m0 + r + 8 * half;
      float val = acc[j][r] + bval;
      if (residual) val += residual[(size_t)m * N + n];
      C[(size_t)m * N + n] = val;
    }
  }
}

// ---------------------------------------------------------------------------
// RMSNorm: one 256-thread block per row of [S, HID].
// ---------------------------------------------------------------------------
__global__ __launch_bounds__(256) void rmsnorm_kernel(
    const float* __restrict__ x, const float* __restrict__ w,
    float* __restrict__ out) {
  const int row = blockIdx.x;
  const float* xr = x + (size_t)row * HID;
  float ss = 0.0f;
  for (int i = threadIdx.x; i < HID; i += blockDim.x) {
    const float v = xr[i];
    ss += v * v;
  }
#pragma unroll
  for (int off = 16; off > 0; off >>= 1) ss += __shfl_xor(ss, off, 32);
  __shared__ float red[8];
  __shared__ float total;
  const int wave = threadIdx.x >> 5, lane = threadIdx.x & 31;
  if (lane == 0) red[wave] = ss;
  __syncthreads();
  if (wave == 0) {
    float v = (lane < 8) ? red[lane] : 0.0f;
#pragma unroll
    for (int off = 4; off > 0; off >>= 1) v += __shfl_xor(v, off, 32);
    if (lane == 0) total = v;
  }
  __syncthreads();
  const float inv = rsqrtf(total / (float)HID + EPS_F);
  for (int i = threadIdx.x; i < HID; i += blockDim.x)
    out[(size_t)row * HID + i] = w[i] * xr[i] * inv;
}

// ---------------------------------------------------------------------------
// Block-mean pooling: x[S, H, HD] -> out[QB, H, HD] (mean over BS rows).
// ---------------------------------------------------------------------------
__global__ __launch_bounds__(128) void pool_kernel(
    const float* __restrict__ x, float* __restrict__ out, int H) {
  const int qb = blockIdx.x / H;
  const int h  = blockIdx.x % H;
  const int d  = threadIdx.x;  // HD threads
  float s = 0.0f;
  for (int i = 0; i < BS; ++i)
    s += x[((size_t)(qb * BS + i) * H + h) * HD + d];
  out[((size_t)qb * H + h) * HD + d] = s * (1.0f / BS);
}

// ---------------------------------------------------------------------------
// Gate scores + softmax + keep mask. One wave32 per (h, qb); lane == kb.
// ---------------------------------------------------------------------------
__global__ __launch_bounds__(32) void gate_kernel(
    const float* __restrict__ qg, const float* __restrict__ kg,
    int* __restrict__ keep) {
  const int h  = blockIdx.x / QB;
  const int qb = blockIdx.x % QB;
  const int kvh = h / GRP;
  const int kb = threadIdx.x;  // 0..31 == QB
  float dot = 0.0f;
  const float* qv = qg + ((size_t)qb * HQ + h) * GH;
  const float* kv = kg + ((size_t)kb * HKV + kvh) * GH;
  for (int g = 0; g < GH; ++g) dot += qv[g] * kv[g];
  const bool causal = (kb <= qb);
  float s = causal ? dot * rsqrtf((float)GH) : NEG_F;
  float mx = s;
#pragma unroll
  for (int off = 16; off > 0; off >>= 1) mx = fmaxf(mx, __shfl_xor(mx, off, 32));
  const float e = __expf(s - mx);
  float sum = e;
#pragma unroll
  for (int off = 16; off > 0; off >>= 1) sum += __shfl_xor(sum, off, 32);
  const float gate = e / sum;
  keep[((size_t)h * QB + qb) * QB + kb] =
      (causal && (gate >= THR_F || kb == qb)) ? 1 : 0;
}

// ---------------------------------------------------------------------------
// RoPE in place on q[S,HQ,HD] and k[S,HKV,HD]; cos/sin are [S, HD].
// ---------------------------------------------------------------------------
__global__ __launch_bounds__(256) void rope_kernel(
    float* __restrict__ q, float* __restrict__ k,
    const float* __restrict__ cosb, const float* __restrict__ sinb) {
  const int s = blockIdx.x;
  const float* cs = cosb + (size_t)s * HD;
  const float* sn = sinb + (size_t)s * HD;
  for (int idx = threadIdx.x; idx < (HQ + HKV) * (HD / 2); idx += blockDim.x) {
    const int hh = idx / (HD / 2);
    const int d  = idx % (HD / 2);
    float* base = (hh < HQ) ? (q + ((size_t)s * HQ + hh) * HD)
                            : (k + ((size_t)s * HKV + (hh - HQ)) * HD);
    const float x1 = base[d], x2 = base[d + HD / 2];
    base[d]          = x1 * cs[d]          - x2 * sn[d];
    base[d + HD / 2] = x2 * cs[d + HD / 2] + x1 * sn[d + HD / 2];
  }
}

// ---------------------------------------------------------------------------
// Block-sparse flash attention. Grid (QB, HQ); 128 threads = 4 waves.
// Q (once) and K/V (per kept key block) are async-copied into LDS and shared
// by all 4 waves. Wave w owns query rows [qb*64 + 16w, +16). Scores and P@V
// via v_wmma_f32_16x16x4_f32; P transposes C-layout -> A-layout through LDS.
// ---------------------------------------------------------------------------
__global__ __launch_bounds__(128) void attn_flash_kernel(
    const float* __restrict__ q, const float* __restrict__ k,
    const float* __restrict__ v, const int* __restrict__ keep,
    float* __restrict__ o) {
  const int qb  = blockIdx.x;
  const int h   = blockIdx.y;
  const int kvh = h / GRP;
  const int tid  = threadIdx.x;
  const int wave = tid >> 5;
  const int lane = tid & 31;
  const int half = lane >> 4;
  const int col  = lane & 15;
  const int m0 = qb * BS + wave * 16;

  __shared__ float Qt[BS * HD];      // 32 KB
  __shared__ float Kt[BS * HD];      // 32 KB
  __shared__ float Vt[BS * HD];      // 32 KB
  __shared__ float Pt[4][16 * 64];   // 16 KB
  float* pw = &Pt[wave][0];
  const unsigned qt_off = lds_off(&Qt[0]);
  const unsigned kt_off = lds_off(&Kt[0]);
  const unsigned vt_off = lds_off(&Vt[0]);

  // ---- stage Q tile (64 x 128) once ----
#pragma unroll
  for (int i = 0; i < 16; ++i) {
    const int e = (tid + i * 128) * 4;
    const int row = e >> 7, d = e & 127;
    async_b128(qt_off + e * 4, q + ((size_t)(qb * BS + row) * HQ + h) * HD + d);
  }
  wait_asynccnt_0();
  __syncthreads();

  v8f zero = {};
  v8f oacc[8];
#pragma unroll
  for (int j = 0; j < 8; ++j) oacc[j] = zero;
  float m_row[8], l_row[8];
#pragma unroll
  for (int r = 0; r < 8; ++r) { m_row[r] = NEG_F; l_row[r] = 0.0f; }

  const float scl = 0.08838834764831845f;  // 1/sqrt(128)

  for (int kb = 0; kb <= qb; ++kb) {
    if (!keep[((size_t)h * QB + qb) * QB + kb]) continue;  // uniform skip

    // ---- stage K/V tiles (64 x 128 each) ----
#pragma unroll
    for (int i = 0; i < 16; ++i) {
      const int e = (tid + i * 128) * 4;
      const int row = e >> 7, d = e & 127;
      const size_t kvoff = ((size_t)(kb * BS + row) * HKV + kvh) * HD + d;
      async_b128(kt_off + e * 4, k + kvoff);
      async_b128(vt_off + e * 4, v + kvoff);
    }
    wait_asynccnt_0();
    __syncthreads();

    // ---- S = Q (16x128) @ K^T (128x64) ----
    v8f sacc[4];
#pragma unroll
    for (int j = 0; j < 4; ++j) sacc[j] = zero;
    const float* Qrow = &Qt[(wave * 16 + col) * HD];
    for (int k0 = 0; k0 < HD; k0 += 4) {
      v2f a;
      a.x = Qrow[k0 + 2 * half];
      a.y = Qrow[k0 + 2 * half + 1];
#pragma unroll
      for (int j = 0; j < 4; ++j) {
        v2f b;
        b.x = Kt[(j * 16 + col) * HD + k0 + 2 * half];
        b.y = Kt[(j * 16 + col) * HD + k0 + 2 * half + 1];
        sacc[j] = wmma4_f32(a, b, sacc[j]);
      }
    }

    // ---- mask + online softmax update ----
#pragma unroll
    for (int r = 0; r < 8; ++r) {
      const int qrow = m0 + r + 8 * half;
      float rmax = NEG_F;
#pragma unroll
      for (int j = 0; j < 4; ++j) {
        float sv = sacc[j][r] * scl;
        const int key = kb * BS + j * 16 + col;
        if (kb == qb && key > qrow) sv = NEG_F;
        sacc[j][r] = sv;
        rmax = fmaxf(rmax, sv);
      }
#pragma unroll
      for (int off = 8; off > 0; off >>= 1)
        rmax = fmaxf(rmax, __shfl_xor(rmax, off, 32));
      const float mnew = fmaxf(m_row[r], rmax);
      const float alpha = __expf(m_row[r] - mnew);
      float rsum = 0.0f;
#pragma unroll
      for (int j = 0; j < 4; ++j) {
        const float p = __expf(sacc[j][r] - mnew);
        pw[(r + 8 * half) * 64 + j * 16 + col] = p;  // ds_store (C-layout)
        rsum += p;
      }
#pragma unroll
      for (int off = 8; off > 0; off >>= 1)
        rsum += __shfl_xor(rsum, off, 32);
      l_row[r] = l_row[r] * alpha + rsum;
      m_row[r] = mnew;
#pragma unroll
      for (int j = 0; j < 8; ++j) oacc[j][r] = oacc[j][r] * alpha;
    }

    // ---- O += P (16x64) @ V (64x128) ----
    for (int k0 = 0; k0 < BS; k0 += 4) {
      v2f a;
      a.x = pw[col * 64 + k0 + 2 * half];  // ds_load (A-layout)
      a.y = pw[col * 64 + k0 + 2 * half + 1];
#pragma unroll
      for (int j = 0; j < 8; ++j) {
        v2f b;
        b.x = Vt[(k0 + 2 * half) * HD + j * 16 + col];
        b.y = Vt[(k0 + 2 * half + 1) * HD + j * 16 + col];
        oacc[j] = wmma4_f32(a, b, oacc[j]);
      }
    }
    __syncthreads();  // done with Kt/Vt before next key block overwrites
  }

#pragma unroll
  for (int r = 0; r < 8; ++r) {
    const float inv = 1.0f / l_row[r];
    const int qrow = m0 + r + 8 * half;
#pragma unroll
    for (int j = 0; j < 8; ++j)
      o[((size_t)qrow * HQ + h) * HD + j * 16 + col] = oacc[j][r] * inv;
  }
}

// ---------------------------------------------------------------------------
// act = silu(gate) * up
// ---------------------------------------------------------------------------
__global__ __launch_bounds__(256) void silu_mul_kernel(
    const float* __restrict__ g, const float* __restrict__ u,
    float* __restrict__ out, size_t n) {
  const size_t i = (size_t)blockIdx.x * blockDim.x + threadIdx.x;
  if (i < n) {
    const float x = g[i];
    out[i] = (x / (1.0f + __expf(-x))) * u[i];
  }
}

// ---------------------------------------------------------------------------
extern "C" void kernel_launch(void* const* d_in, const int* in_sizes, int n_in,
                              void* d_out, int out_size, void* d_ws, size_t ws_size,
                              hipStream_t stream) {
  (void)in_sizes; (void)n_in; (void)out_size; (void)ws_size;
  const float* hidden = (const float*)d_in[0];
  const float* cosb   = (const float*)d_in[1];
  const float* sinb   = (const float*)d_in[2];
  const float* ln1    = (const float*)d_in[3];
  const float* ln2    = (const float*)d_in[4];
  const float* Wq     = (const float*)d_in[5];
  const float* bq     = (const float*)d_in[6];
  const float* Wk     = (const float*)d_in[7];
  const float* bk     = (const float*)d_in[8];
  const float* Wv     = (const float*)d_in[9];
  const float* bv     = (const float*)d_in[10];
  const float* Wo     = (const float*)d_in[11];
  const float* gWq    = (const float*)d_in[12];
  const float* gWk    = (const float*)d_in[13];
  const float* Wgate  = (const float*)d_in[14];
  const float* Wup    = (const float*)d_in[15];
  const float* Wdown  = (const float*)d_in[16];
  float* out = (float*)d_out;

  float* ws = (float*)d_ws;
  size_t off = 0;
  float* h1    = ws + off; off += (size_t)S_LEN * HID;
  float* qbuf  = ws + off; off += (size_t)S_LEN * HQ * HD;
  float* kbuf  = ws + off; off += (size_t)S_LEN * HKV * HD;
  float* vbuf  = ws + off; off += (size_t)S_LEN * HKV * HD;
  float* obuf  = ws + off; off += (size_t)S_LEN * HQ * HD;
  float* h2    = ws + off; off += (size_t)S_LEN * HID;
  float* h3    = ws + off; off += (size_t)S_LEN * HID;
  float* qpool = ws + off; off += (size_t)QB * HQ * HD;
  float* kpool = ws + off; off += (size_t)QB * HKV * HD;
  float* qg    = ws + off; off += (size_t)QB * HQ * GH;
  float* kg    = ws + off; off += (size_t)QB * HKV * GH;
  int*   keep  = (int*)(ws + off); off += (size_t)HQ * QB * QB;
  float* gbuf  = ws + off; off += (size_t)S_LEN * INTER;
  float* ubuf  = ws + off; off += (size_t)S_LEN * INTER;

  // 1) pre-attention RMSNorm
  rmsnorm_kernel<<<S_LEN, 256, 0, stream>>>(hidden, ln1, h1);
  // 2) QKV projections (fused bias)
  gemm_wmma_f32_kernel<<<dim3(HQ * HD / BN, S_LEN / BM), 256, 0, stream>>>(
      h1, Wq, bq, nullptr, qbuf, S_LEN, HQ * HD, HID);
  gemm_wmma_f32_kernel<<<dim3(HKV * HD / BN, S_LEN / BM), 256, 0, stream>>>(
      h1, Wk, bk, nullptr, kbuf, S_LEN, HKV * HD, HID);
  gemm_wmma_f32_kernel<<<dim3(HKV * HD / BN, S_LEN / BM), 256, 0, stream>>>(
      h1, Wv, bv, nullptr, vbuf, S_LEN, HKV * HD, HID);
  // 3) gate path: pool (mean commutes with linear) -> small WMMA GEMMs -> mask
  pool_kernel<<<QB * HQ, HD, 0, stream>>>(qbuf, qpool, HQ);
  pool_kernel<<<QB * HKV, HD, 0, stream>>>(kbuf, kpool, HKV);
  gemm_wmma_f32_kernel<<<dim3(GH / BN, QB * HQ / BM), 256, 0, stream>>>(
      qpool, gWq, nullptr, nullptr, qg, QB * HQ, GH, HD);
  gemm_wmma_f32_kernel<<<dim3(GH / BN, QB * HKV / BM), 256, 0, stream>>>(
      kpool, gWk, nullptr, nullptr, kg, QB * HKV, GH, HD);
  gate_kernel<<<HQ * QB, 32, 0, stream>>>(qg, kg, keep);
  // 4) RoPE (in place), then block-sparse flash attention
  rope_kernel<<<S_LEN, 256, 0, stream>>>(qbuf, kbuf, cosb, sinb);
  attn_flash_kernel<<<dim3(QB, HQ), 128, 0, stream>>>(qbuf, kbuf, vbuf, keep, obuf);
  // 5) output projection + residual
  gemm_wmma_f32_kernel<<<dim3(HID / BN, S_LEN / BM), 256, 0, stream>>>(
      obuf, Wo, nullptr, hidden, h2, S_LEN, HID, HQ * HD);
  // 6) MLP: norm -> gate/up -> silu*up -> down + residual (writes d_out)
  rmsnorm_kernel<<<S_LEN, 256, 0, stream>>>(h2, ln2, h3);
  gemm_wmma_f32_kernel<<<dim3(INTER / BN, S_LEN / BM), 256, 0, stream>>>(
      h3, Wgate, nullptr, nullptr, gbuf, S_LEN, INTER, HID);
  gemm_wmma_f32_kernel<<<dim3(INTER / BN, S_LEN / BM), 256, 0, stream>>>(
      h3, Wup, nullptr, nullptr, ubuf, S_LEN, INTER, HID);
  const size_t n = (size_t)S_LEN * INTER;
  silu_mul_kernel<<<(unsigned)((n + 255) / 256), 256, 0, stream>>>(gbuf, ubuf, gbuf, n);
  gemm_wmma_f32_kernel<<<dim3(HID / BN, S_LEN / BM), 256, 0, stream>>>(
      gbuf, Wdown, nullptr, h2, out, S_LEN, HID, INTER);
}